// Sampler_17162689315312
// MI455X (gfx1250) — compile-verified
//
#include <hip/hip_runtime.h>
#include <stdint.h>

// Problem shape (fixed by the reference): B=128 rows, V=128000 vocab.
#define BDIM   1024
#define NWAVE  (BDIM / 32)
#define NBINS  2048
#define CAP    4096
#define ROWV   128000
#define NF4    (ROWV / 4)        // 32000 float4 per row
#define NSEG   (ROWV / 1024)     // 125 segments of 1024 elements
#define FULLI  31                // 31 full async-staged tiles of 1024 float4
#define TAILN  (NF4 - FULLI * BDIM)  // 256 float4 tail
#define FPSCALE 1073741824.0f    // 2^30 fixed-point scale for deterministic sums

// ---------- float <-> descending-orderable key ----------
static __device__ __forceinline__ uint32_t f2key(float f) {
    uint32_t u = __float_as_uint(f);
    return (u & 0x80000000u) ? ~u : (u | 0x80000000u);
}
static __device__ __forceinline__ float key2f(uint32_t k) {
    uint32_t u = (k & 0x80000000u) ? (k ^ 0x80000000u) : ~k;
    return __uint_as_float(u);
}

// ---------- gfx1250 async global->LDS copy (tracked by ASYNCcnt) ----------
static __device__ __forceinline__ void async_copy_b128(uint32_t lds_byte_addr, uint64_t gaddr) {
    asm volatile("global_load_async_to_lds_b128 %0, %1, off"
                 :: "v"(lds_byte_addr), "v"(gaddr) : "memory");
}
static __device__ __forceinline__ void wait_async_le1() {
    asm volatile("s_wait_asynccnt 0x1" ::: "memory");
}
static __device__ __forceinline__ void wait_async_0() {
    asm volatile("s_wait_asynccnt 0x0" ::: "memory");
}

// ---------- block max (1024 threads, wave32) ----------
static __device__ float blockMax(float v, float* s_red) {
    const int lane = threadIdx.x & 31, wv = threadIdx.x >> 5;
    for (int o = 16; o > 0; o >>= 1) v = fmaxf(v, __shfl_down(v, o, 32));
    __syncthreads();
    if (lane == 0) s_red[wv] = v;
    __syncthreads();
    if (wv == 0) {
        float x = s_red[lane];
        for (int o = 16; o > 0; o >>= 1) x = fmaxf(x, __shfl_down(x, o, 32));
        if (lane == 0) s_red[0] = x;
    }
    __syncthreads();
    float r = s_red[0];
    __syncthreads();
    return r;
}

// ---------- descending-order transition finder over integer histogram ----------
// Finds bin b with sumAbove(b) < target <= sumAbove(b)+hist[b] (sums over bins > b).
// Bin values are exact integers in double (< 2^53), so the suffix scan is exact
// and deterministic. Writes *selBin, *selAbove; optionally *totalOut.
template <typename TB>
static __device__ void findTransDesc(const TB* hist, int nbins, double target,
                                     double* s_dred, int* selBin, double* selAbove,
                                     double* totalOut) {
    const int tid = threadIdx.x, lane = tid & 31, wv = tid >> 5;
    __syncthreads();
    if (tid == 0) { *selBin = 0; *selAbove = 0.0; }
    const int items = nbins >> 10;   // 1 or 2 (BDIM == 1024)
    double loc[2] = {0.0, 0.0};
    double lsum = 0.0;
    for (int j = 0; j < items; ++j) {
        const int p = tid * items + j;
        loc[j] = (double)hist[nbins - 1 - p];
        lsum += loc[j];
    }
    double inc = lsum;
    for (int o = 1; o < 32; o <<= 1) { double n = __shfl_up(inc, o, 32); if (lane >= o) inc += n; }
    if (lane == 31) s_dred[wv] = inc;
    __syncthreads();
    if (wv == 0) {
        double x = s_dred[lane];
        for (int o = 1; o < 32; o <<= 1) { double n = __shfl_up(x, o, 32); if (lane >= o) x += n; }
        s_dred[lane] = x;
    }
    __syncthreads();
    if (totalOut && tid == 0) *totalOut = s_dred[NWAVE - 1];
    double excl = ((wv > 0) ? s_dred[wv - 1] : 0.0) + (inc - lsum);
    for (int j = 0; j < items; ++j) {
        const double e1 = excl + loc[j];
        if (excl < target && target <= e1) {
            *selBin = nbins - 1 - (tid * items + j);
            *selAbove = excl;
        }
        excl = e1;
    }
    __syncthreads();
}

__global__ __launch_bounds__(BDIM) void sampler_kernel(
        const float* __restrict__ logits, const float* __restrict__ temps,
        const int* __restrict__ topks, const float* __restrict__ topps,
        const float* __restrict__ minps, const float* __restrict__ us,
        int* __restrict__ out) {
    // 16 KB bin arena, time-multiplexed:
    //   counts : 2 banked copies of 2048 u32 (bank = wave parity, merged pre-select)
    //   weights: 2048 u64 Q2.30 fixed-point sums (top-p select over LDS survivor list)
    __shared__ __align__(16) unsigned char s_arena[NBINS * 8];
    unsigned*           s_c = (unsigned*)s_arena;
    unsigned long long* s_w = (unsigned long long*)s_arena;

    __shared__ float s_stage[2][BDIM][4];   // 32 KB: async staging; reused as survivor lists
    __shared__ float s_seg[NSEG];           // per-1024-segment masked prob-mass
    __shared__ float s_red[NWAVE];
    __shared__ double s_dred[NWAVE];
    __shared__ int s_icnt[NWAVE];
    __shared__ int s_selBin;      __shared__ double s_selAbove;  __shared__ double s_total;
    __shared__ unsigned s_cnt;
    __shared__ int s_selSeg;      __shared__ float s_before;     __shared__ float s_target;

    float* s_listL = &s_stage[0][0][0];   // CAP survivor logits
    float* s_listE = &s_stage[1][0][0];   // CAP survivor exp terms

    const int row = blockIdx.x;
    const int tid = threadIdx.x;
    const int lane = tid & 31, wv = tid >> 5;
    const unsigned bank = (unsigned)(wv & 1) << 11;   // banked count histogram offset
    const float* rowp = logits + (size_t)row * ROWV;
    const float4* rowv = (const float4*)rowp;

    const float T = temps[row];
    const float invT = 1.0f / T;
    int k = topks[row]; if (k < 1) k = 1; if (k > ROWV) k = ROWV;
    const float top_p = topps[row];
    const float min_p = minps[row];
    const float u = us[row];

    auto clearBins = [&]() {   // clears full arena (both count banks / all u64 bins)
        for (int i = tid; i < 2 * NBINS; i += BDIM) s_c[i] = 0u;
        __syncthreads();
    };
    auto mergeBanks = [&](int nbins) {
        __syncthreads();
        for (int i = tid; i < nbins; i += BDIM) s_c[i] += s_c[NBINS + i];
        __syncthreads();
    };

    // ================= P1: row max + radix round-0 count histogram =================
    // Streamed via gfx1250 async global->LDS copies, per-wave double buffering.
    clearBins();

    float lmax = -INFINITY;
    {
        const uint64_t gbase = (uint64_t)(uintptr_t)rowp;
        const uint32_t lbase = (uint32_t)(uintptr_t)(&s_stage[0][0][0]) + (uint32_t)tid * 16u;
        auto acc0 = [&](float x) {
            lmax = fmaxf(lmax, x);
            atomicAdd(&s_c[bank + (f2key(x) >> 21)], 1u);
        };
        async_copy_b128(lbase,          gbase + (uint64_t)tid * 16u);
        async_copy_b128(lbase + 16384u, gbase + (uint64_t)(BDIM + tid) * 16u);
        for (int i = 0; i < FULLI; ++i) {
            if (i == FULLI - 1) wait_async_0(); else wait_async_le1();
            float4 v = *(const float4*)(&s_stage[i & 1][tid][0]);
            acc0(v.x); acc0(v.y); acc0(v.z); acc0(v.w);
            if (i + 2 < FULLI)
                async_copy_b128(lbase + (uint32_t)(i & 1) * 16384u,
                                gbase + (uint64_t)((i + 2) * BDIM + tid) * 16u);
        }
        if (tid < TAILN) {   // 256-float4 tail, direct load
            float4 v = rowv[FULLI * BDIM + tid];
            acc0(v.x); acc0(v.y); acc0(v.z); acc0(v.w);
        }
    }
    __syncthreads();
    lmax = blockMax(lmax, s_red);

    // ================= top-k radix select (11 / 11 / 10 bits) =================
    double tk = (double)k;
    mergeBanks(NBINS);
    findTransDesc<unsigned>(s_c, NBINS, tk, s_dred, &s_selBin, &s_selAbove, nullptr);
    const uint32_t b0 = (uint32_t)s_selBin;  tk -= s_selAbove;
    __syncthreads();

    clearBins();
    for (int it = 0; it <= FULLI; ++it) {            // P2
        const int i4 = tid + it * BDIM;
        if (i4 + BDIM < NF4) __builtin_prefetch(rowv + i4 + BDIM, 0, 0);
        if (i4 < NF4) {
            float4 v = rowv[i4];
            auto f = [&](float x) {
                uint32_t key = f2key(x);
                if ((key >> 21) == b0) atomicAdd(&s_c[bank + ((key >> 10) & 0x7FFu)], 1u);
            };
            f(v.x); f(v.y); f(v.z); f(v.w);
        }
    }
    mergeBanks(NBINS);
    findTransDesc<unsigned>(s_c, NBINS, tk, s_dred, &s_selBin, &s_selAbove, nullptr);
    const uint32_t b1 = (uint32_t)s_selBin;  tk -= s_selAbove;
    const uint32_t pfx22 = (b0 << 11) | b1;
    __syncthreads();

    clearBins();
    for (int it = 0; it <= FULLI; ++it) {            // P3
        const int i4 = tid + it * BDIM;
        if (i4 + BDIM < NF4) __builtin_prefetch(rowv + i4 + BDIM, 0, 0);
        if (i4 < NF4) {
            float4 v = rowv[i4];
            auto f = [&](float x) {
                uint32_t key = f2key(x);
                if ((key >> 10) == pfx22) atomicAdd(&s_c[bank + (key & 0x3FFu)], 1u);
            };
            f(v.x); f(v.y); f(v.z); f(v.w);
        }
    }
    mergeBanks(1024);
    findTransDesc<unsigned>(s_c, 1024, tk, s_dred, &s_selBin, &s_selAbove, nullptr);
    const uint32_t Kk = (b0 << 21) | (b1 << 10) | (uint32_t)s_selBin;   // exact k-th key
    __syncthreads();

    // ================= P4: compact survivors (key >= Kk) into LDS =================
    if (tid == 0) s_cnt = 0u;
    __syncthreads();
    for (int it = 0; it <= FULLI; ++it) {
        const int i4 = tid + it * BDIM;
        if (i4 < NF4) {
            float4 v = rowv[i4];
            auto f = [&](float x) {
                if (f2key(x) >= Kk) {
                    unsigned id = atomicAdd(&s_cnt, 1u);
                    if (id < CAP) { s_listL[id] = x; s_listE[id] = __expf((x - lmax) * invT); }
                }
            };
            f(v.x); f(v.y); f(v.z); f(v.w);
        }
    }
    __syncthreads();
    const int C = (s_cnt < (unsigned)CAP) ? (int)s_cnt : CAP;

    // ========= top-p: weighted radix select over the LDS survivor list =========
    // Q2.30 fixed-point weights in u64 bins -> integer atomics -> deterministic.
    clearBins();
    for (int i = tid; i < C; i += BDIM) {
        unsigned long long w = (unsigned long long)(s_listE[i] * FPSCALE);
        atomicAdd(&s_w[f2key(s_listL[i]) >> 21], w);
    }
    __syncthreads();
    findTransDesc<unsigned long long>(s_w, NBINS, -1.0, s_dred,
                                      &s_selBin, &s_selAbove, &s_total);  // total only
    double ptarget = (double)top_p * s_total;
    __syncthreads();
    findTransDesc<unsigned long long>(s_w, NBINS, ptarget, s_dred,
                                      &s_selBin, &s_selAbove, nullptr);
    const uint32_t p0 = (uint32_t)s_selBin;  ptarget -= s_selAbove;
    __syncthreads();

    clearBins();
    for (int i = tid; i < C; i += BDIM) {
        uint32_t key = f2key(s_listL[i]);
        if ((key >> 21) == p0)
            atomicAdd(&s_w[(key >> 10) & 0x7FFu],
                      (unsigned long long)(s_listE[i] * FPSCALE));
    }
    __syncthreads();
    findTransDesc<unsigned long long>(s_w, NBINS, ptarget, s_dred,
                                      &s_selBin, &s_selAbove, nullptr);
    const uint32_t p1 = (uint32_t)s_selBin;  ptarget -= s_selAbove;
    const uint32_t ppfx22 = (p0 << 11) | p1;
    __syncthreads();

    clearBins();
    for (int i = tid; i < C; i += BDIM) {
        uint32_t key = f2key(s_listL[i]);
        if ((key >> 10) == ppfx22)
            atomicAdd(&s_w[key & 0x3FFu],
                      (unsigned long long)(s_listE[i] * FPSCALE));
    }
    __syncthreads();
    findTransDesc<unsigned long long>(s_w, 1024, ptarget, s_dred,
                                      &s_selBin, &s_selAbove, nullptr);
    const uint32_t Kp = (p0 << 21) | (p1 << 10) | (uint32_t)s_selBin;
    const float t_pf = key2f(Kp);
    __syncthreads();

    // min-p threshold is analytic: max surviving prob term is exp(0)=1.
    const float t_m = (min_p > 0.0f) ? (lmax + T * __logf(min_p)) : -INFINITY;
    const float t_fin = fmaxf(t_pf, t_m);

    // ============ P5: deterministic per-segment masked mass (wave-owned) ============
    for (int s = wv; s < NSEG; s += NWAVE) {
        float ls = 0.0f;
        for (int j = 0; j < 8; ++j) {
            float4 v = rowv[s * 256 + j * 32 + lane];
            ls += (v.x >= t_fin) ? __expf((v.x - lmax) * invT) : 0.0f;
            ls += (v.y >= t_fin) ? __expf((v.y - lmax) * invT) : 0.0f;
            ls += (v.z >= t_fin) ? __expf((v.z - lmax) * invT) : 0.0f;
            ls += (v.w >= t_fin) ? __expf((v.w - lmax) * invT) : 0.0f;
        }
        for (int o = 16; o > 0; o >>= 1) ls += __shfl_down(ls, o, 32);
        if (lane == 0) s_seg[s] = ls;
    }
    __syncthreads();

    // Serial, self-consistent crossing search over 125 segments (thread 0).
    if (tid == 0) {
        float tot = 0.0f;
        for (int s = 0; s < NSEG; ++s) tot += s_seg[s];
        const float tgt = u * tot;
        int sel = -2; float bef = 0.0f;
        if (tgt <= 0.0f) sel = -3;
        else {
            float run = 0.0f;
            for (int s = 0; s < NSEG; ++s) {
                const float nx = run + s_seg[s];
                if (run < tgt && tgt <= nx) { sel = s; bef = run; break; }
                run = nx;
            }
        }
        s_selSeg = sel; s_before = bef; s_target = tgt;
    }
    __syncthreads();
    const int selSeg = s_selSeg;
    const float tgt = s_target, before = s_before;

    // ================= P6: localize token inside the selected segment =================
    if (selSeg < 0) {
        if (tid == 0) out[row] = (selSeg == -3) ? 0 : (ROWV - 1);
    } else {
        const int base = selSeg * 1024;
        const float x = rowp[base + tid];
        float e = (x >= t_fin) ? __expf((x - lmax) * invT) : 0.0f;
        float inc = e;
        for (int o = 1; o < 32; o <<= 1) { float n = __shfl_up(inc, o, 32); if (lane >= o) inc += n; }
        __syncthreads();
        if (lane == 31) s_red[wv] = inc;
        __syncthreads();
        if (wv == 0) {
            float t0 = s_red[lane];
            for (int o = 1; o < 32; o <<= 1) { float n = __shfl_up(t0, o, 32); if (lane >= o) t0 += n; }
            s_red[lane] = t0;
        }
        __syncthreads();
        inc += (wv > 0) ? s_red[wv - 1] : 0.0f;
        const int pred = (before + inc < tgt) ? 1 : 0;
        const unsigned long long bal = __ballot(pred);
        if (lane == 0) s_icnt[wv] = __popcll(bal);
        __syncthreads();
        if (tid == 0) {
            int c = 0;
            for (int i = 0; i < NWAVE; ++i) c += s_icnt[i];
            int token = base + c;
            if (token > ROWV - 1) token = ROWV - 1;
            out[row] = token;
        }
    }
}

extern "C" void kernel_launch(void* const* d_in, const int* in_sizes, int n_in,
                              void* d_out, int out_size, void* d_ws, size_t ws_size,
                              hipStream_t stream) {
    (void)n_in; (void)out_size; (void)d_ws; (void)ws_size;
    const float* logits = (const float*)d_in[0];
    const float* temps  = (const float*)d_in[1];
    const int*   topks  = (const int*)d_in[2];
    const float* topps  = (const float*)d_in[3];
    const float* minps  = (const float*)d_in[4];
    const float* us     = (const float*)d_in[5];
    int* out = (int*)d_out;
    const int B = in_sizes[1];   // 128 rows, one workgroup (32 wave32 waves) per row
    sampler_kernel<<<B, BDIM, 0, stream>>>(logits, temps, topks, topps, minps, us, out);
}